// NeuralSCM_38663295599062
// MI455X (gfx1250) — compile-verified
//
#include <hip/hip_runtime.h>
#include <stdint.h>

// ---------------------------------------------------------------------------
// NeuralSCM fused kernel for MI455X (gfx1250), fp32 WMMA path.
//   D = A(16x4 f32) x B(4x16 f32) + C(16x16 f32)  via v_wmma_f32_16x16x4_f32
// Weights prefetched with ASYNCcnt-tracked global->LDS async copies that
// perform the K/N transpose in flight (per-lane LDS addressing), so the hot
// B-fragment loads stay single ds_load_2addr_b64 with bank-conflict-free
// strides. W2 is double-buffered; node i+1's weights stream during node i.
// One workgroup (8 wave32) carries 128 batch rows through all 8 DAG nodes.
// ---------------------------------------------------------------------------

typedef float v2f __attribute__((ext_vector_type(2)));
typedef float v8f __attribute__((ext_vector_type(8)));

#define SCM_BATCH 65536
#define SCM_HID   128
#define SW2_K 132          // W2^T row stride (dwords): banks 4n / 4n+2 -> conflict-free
#define SW1_K 26           // W1^T row stride
#define SW3_K 132          // W3^T row stride
#define SH_STRIDE 20       // per-wave h slot stride: rows*20 mod 64 all distinct

struct SCMParams {
  const float* u;                      // [8][B][8]
  const float* W1[8]; const float* b1[8];
  const float* W2[8]; const float* b2[8];
  const float* W3[8]; const float* b3[8];
};

__device__ __forceinline__ v8f wmma_f32_k4(v2f a, v2f b, v8f c) {
  return __builtin_amdgcn_wmma_f32_16x16x4_f32(
      /*neg_a=*/false, a, /*neg_b=*/false, b,
      /*c_mod=*/(short)0, c, /*reuse_a=*/false, /*reuse_b=*/false);
}

// Low 32 bits of a generic pointer to LDS = byte offset within the wave's LDS.
__device__ __forceinline__ uint32_t lds_off(const void* p) {
  return (uint32_t)(uintptr_t)p;
}

// Async global->LDS copy, 4B per lane, per-lane LDS destination (transposing
// scatter), tracked by ASYNCcnt. Global side stays coalesced.
__device__ __forceinline__ void async_b32(uint32_t lds_byte, const float* gbase,
                                          uint32_t gbyte) {
  asm volatile("global_load_async_to_lds_b32 %0, %1, %2"
               :: "v"(lds_byte), "v"(gbyte), "s"(gbase)
               : "memory");
}

// Async ops complete in order: after fully issuing batch i+1, waiting to depth
// 32 guarantees every op of batch i has completed.
__device__ __forceinline__ void async_wait0(void) {
  asm volatile("s_wait_asynccnt 0x0" ::: "memory");
}
__device__ __forceinline__ void async_wait32(void) {
  asm volatile("s_wait_asynccnt 0x20" ::: "memory");
}

// W2 [128k][128n] -> LDS W2^T [n][SW2_K] ; 64 async b32 per thread.
template <int NODE>
__device__ __forceinline__ void stage_w2_async(const SCMParams& P, float* sW2T, int tid) {
  const float* g = P.W2[NODE];
  const uint32_t o = lds_off(sW2T);
#pragma unroll
  for (int j = 0; j < 64; ++j) {
    const uint32_t idx = (uint32_t)tid + 256u * (uint32_t)j;   // k*128+n
    const uint32_t k = idx >> 7, n = idx & 127u;
    async_b32(o + (n * SW1_K * 0u + n * SW2_K + k) * 4u, g, idx * 4u);
  }
}

// W1 [INk][128n] -> W1^T [n][SW1_K]; W3 [128k][8n] -> W3^T [n][SW3_K].
template <int NODE, int NPAR>
__device__ __forceinline__ void stage_w13_async(const SCMParams& P, float* sW1T,
                                                float* sW3T, int tid) {
  constexpr int IN = 8 * (1 + NPAR);
  const float* g1 = P.W1[NODE];
  const float* g3 = P.W3[NODE];
  const uint32_t o1 = lds_off(sW1T), o3 = lds_off(sW3T);
#pragma unroll
  for (int j = 0; j < IN / 2; ++j) {
    const uint32_t idx = (uint32_t)tid + 256u * (uint32_t)j;   // k*128+n
    const uint32_t k = idx >> 7, n = idx & 127u;
    async_b32(o1 + (n * SW1_K + k) * 4u, g1, idx * 4u);
  }
#pragma unroll
  for (int j = 0; j < 4; ++j) {
    const uint32_t idx = (uint32_t)tid + 256u * (uint32_t)j;   // k*8+n
    const uint32_t k = idx >> 3, n = idx & 7u;
    async_b32(o3 + (n * SW3_K + k) * 4u, g3, idx * 4u);
  }
}

template <int NODE, int NPAR, int P0, int P1>
__device__ __forceinline__ void compute_node(const SCMParams& P, float* __restrict__ out,
                                             const float* sW1T, const float* sW2T,
                                             const float* sW3T, float* sH,
                                             int wave, int lane) {
  constexpr int IN = 8 * (1 + NPAR);   // 8, 16, or 24
  constexpr int NF = IN / 4;           // layer-1 K steps
  const int laneM  = lane & 15;
  const int laneHi = lane >> 4;
  const int rowBase = blockIdx.x * 128 + wave * 16;
  const int hbase   = wave * 16 * SH_STRIDE;

  const float* gB1 = P.b1[NODE];
  const float* gB2 = P.b2[NODE];
  const float* gB3 = P.b3[NODE];

  // ---- layer-1 A fragments (x = [u_i | z_p0 | z_p1]) in registers ----
  v2f xf[NF];
  {
    const int row = rowBase + laneM;
#pragma unroll
    for (int j = 0; j < NF; ++j) {
      const int c = 4 * j + 2 * laneHi;
      const float* src;
      if (c < 8) {
        src = P.u + ((size_t)NODE * SCM_BATCH + row) * 8 + c;
      } else {
        const int p = (((c - 8) >> 3) != 0) ? P1 : P0;
        src = out + ((size_t)p * SCM_BATCH + row) * 8 + ((c - 8) & 7);
      }
      xf[j] = *(const v2f*)src;
    }
  }

  v8f h2[8];
#pragma unroll
  for (int nt = 0; nt < 8; ++nt) h2[nt] = (v8f){};

  // ---- layers 1+2 fused: h1 produced per 16-col chunk, consumed as K chunk ----
#pragma unroll
  for (int c = 0; c < 8; ++c) {
    v8f a1 = (v8f){};
#pragma unroll
    for (int j = 0; j < NF; ++j) {
      v2f b = *(const v2f*)&sW1T[(c * 16 + laneM) * SW1_K + 4 * j + 2 * laneHi];
      a1 = wmma_f32_k4(xf[j], b, a1);
    }
    const float bv = gB1[c * 16 + laneM];
#pragma unroll
    for (int r = 0; r < 8; ++r) {
      float v = a1[r] + bv;
      v = v > 0.f ? v : 0.f;
      sH[hbase + (r + 8 * laneHi) * SH_STRIDE + laneM] = v;   // C/D -> LDS
    }
#pragma unroll
    for (int kk = 0; kk < 4; ++kk) {
      v2f a = *(const v2f*)&sH[hbase + laneM * SH_STRIDE + 4 * kk + 2 * laneHi];
      const int kg = c * 16 + 4 * kk + 2 * laneHi;
#pragma unroll
      for (int nt = 0; nt < 8; ++nt) {
        v2f b = *(const v2f*)&sW2T[(nt * 16 + laneM) * SW2_K + kg];
        h2[nt] = wmma_f32_k4(a, b, h2[nt]);
      }
    }
  }

  // ---- bias2 + relu on register-resident h2 ----
#pragma unroll
  for (int nt = 0; nt < 8; ++nt) {
    const float bv = gB2[nt * 16 + laneM];
#pragma unroll
    for (int r = 0; r < 8; ++r) {
      float v = h2[nt][r] + bv;
      h2[nt][r] = v > 0.f ? v : 0.f;
    }
  }

  // ---- layer 3: z = h2 @ W3 + b3; W3^T has 8 valid rows, lanes n>=8 reuse
  //      row n-8 -> garbage lands only in never-stored columns 8..15. ----
  v8f zacc = (v8f){};
#pragma unroll
  for (int c = 0; c < 8; ++c) {
#pragma unroll
    for (int r = 0; r < 8; ++r)
      sH[hbase + (r + 8 * laneHi) * SH_STRIDE + laneM] = h2[c][r];
#pragma unroll
    for (int kk = 0; kk < 4; ++kk) {
      v2f a = *(const v2f*)&sH[hbase + laneM * SH_STRIDE + 4 * kk + 2 * laneHi];
      const int kg = c * 16 + 4 * kk + 2 * laneHi;
      v2f b = *(const v2f*)&sW3T[(laneM & 7) * SW3_K + kg];
      zacc = wmma_f32_k4(a, b, zacc);
    }
  }
  const float b3v = gB3[laneM & 7];
  if (laneM < 8) {
#pragma unroll
    for (int r = 0; r < 8; ++r)
      out[((size_t)NODE * SCM_BATCH + rowBase + r + 8 * laneHi) * 8 + laneM] =
          zacc[r] + b3v;
  }
}

__global__ __launch_bounds__(256) void scm_kernel(SCMParams P, float* __restrict__ out) {
  __shared__ float sW2T[2][SCM_HID * SW2_K];   // 135168 B: W2^T ping-pong
  __shared__ float sW1T[SCM_HID * SW1_K];      //  13312 B
  __shared__ float sW3T[8 * SW3_K];            //   4224 B
  __shared__ float sH[8 * 16 * SH_STRIDE];     //  10240 B per-wave transpose slots

  const int tid  = threadIdx.x;
  const int wave = tid >> 5;
  const int lane = tid & 31;

  // Topological order: PARENTS = [[],[0],[0],[1,2],[3],[3],[4,5],[6]]
  // Node i computes from sW2T[i&1]; node i+1's W2 streams into sW2T[(i+1)&1]
  // during node i's compute. In-order async completion: wait depth 32 after
  // issuing batch i+1 proves batch i (W13 + W2) has fully landed.

  stage_w13_async<0, 0>(P, sW1T, sW3T, tid);
  stage_w2_async<0>(P, sW2T[0], tid);

  stage_w2_async<1>(P, sW2T[1], tid);
  async_wait32(); __syncthreads();
  compute_node<0, 0, 0, 0>(P, out, sW1T, sW2T[0], sW3T, sH, wave, lane);
  __syncthreads();

  stage_w13_async<1, 1>(P, sW1T, sW3T, tid);
  stage_w2_async<2>(P, sW2T[0], tid);
  async_wait32(); __syncthreads();
  compute_node<1, 1, 0, 0>(P, out, sW1T, sW2T[1], sW3T, sH, wave, lane);
  __syncthreads();

  stage_w13_async<2, 1>(P, sW1T, sW3T, tid);
  stage_w2_async<3>(P, sW2T[1], tid);
  async_wait32(); __syncthreads();
  compute_node<2, 1, 0, 0>(P, out, sW1T, sW2T[0], sW3T, sH, wave, lane);
  __syncthreads();

  stage_w13_async<3, 2>(P, sW1T, sW3T, tid);
  stage_w2_async<4>(P, sW2T[0], tid);
  async_wait32(); __syncthreads();
  compute_node<3, 2, 1, 2>(P, out, sW1T, sW2T[1], sW3T, sH, wave, lane);
  __syncthreads();

  stage_w13_async<4, 1>(P, sW1T, sW3T, tid);
  stage_w2_async<5>(P, sW2T[1], tid);
  async_wait32(); __syncthreads();
  compute_node<4, 1, 3, 0>(P, out, sW1T, sW2T[0], sW3T, sH, wave, lane);
  __syncthreads();

  stage_w13_async<5, 1>(P, sW1T, sW3T, tid);
  stage_w2_async<6>(P, sW2T[0], tid);
  async_wait32(); __syncthreads();
  compute_node<5, 1, 3, 0>(P, out, sW1T, sW2T[1], sW3T, sH, wave, lane);
  __syncthreads();

  stage_w13_async<6, 2>(P, sW1T, sW3T, tid);
  stage_w2_async<7>(P, sW2T[1], tid);
  async_wait32(); __syncthreads();
  compute_node<6, 2, 4, 5>(P, out, sW1T, sW2T[0], sW3T, sH, wave, lane);
  __syncthreads();

  stage_w13_async<7, 1>(P, sW1T, sW3T, tid);
  async_wait0(); __syncthreads();
  compute_node<7, 1, 6, 0>(P, out, sW1T, sW2T[1], sW3T, sH, wave, lane);
}

extern "C" void kernel_launch(void* const* d_in, const int* in_sizes, int n_in,
                              void* d_out, int out_size, void* d_ws, size_t ws_size,
                              hipStream_t stream) {
  (void)in_sizes; (void)n_in; (void)out_size; (void)d_ws; (void)ws_size;
  SCMParams P;
  P.u = (const float*)d_in[0];
  for (int i = 0; i < 8; ++i) {
    P.W1[i] = (const float*)d_in[1 + 6 * i + 0];
    P.b1[i] = (const float*)d_in[1 + 6 * i + 1];
    P.W2[i] = (const float*)d_in[1 + 6 * i + 2];
    P.b2[i] = (const float*)d_in[1 + 6 * i + 3];
    P.W3[i] = (const float*)d_in[1 + 6 * i + 4];
    P.b3[i] = (const float*)d_in[1 + 6 * i + 5];
  }
  dim3 grid(SCM_BATCH / 128);
  dim3 block(256);
  hipLaunchKernelGGL(scm_kernel, grid, block, 0, stream, P, (float*)d_out);
}